// PaiConv_38981123178750
// MI455X (gfx1250) — compile-verified
//
#include <hip/hip_runtime.h>
#include <math.h>

#define Bb   8
#define Nn   2048
#define Cin  64
#define Cout 64
#define Kk   20
#define Ks   20
#define MT   16          // points (M tile) per block
#define CH   8           // channels per chunk
#define NCHUNK 16        // 128 feat rows / CH
#define KCH  (CH*Ks)     // 160 K-values per chunk
#define WROW (2*Cin*Ks)  // 2560 = conv_w row length

typedef float v2f __attribute__((ext_vector_type(2)));
typedef float v8f __attribute__((ext_vector_type(8)));

// ---------------- Kernel 0: pack weights into pair-interleaved transposed layout ----
// wP[(k>>1)*128 + o*2 + (k&1)]  = conv_w[o][k]      (k in [0,2560), o in [0,64))
// moP[(c>>1)*128 + o*2 + (c&1)] = mlp_out_w[o][c]   (c,o in [0,64))
__global__ void __launch_bounds__(256)
pack_weights_kernel(const float* __restrict__ conv_w, const float* __restrict__ mlp_out_w,
                    float* __restrict__ wP, float* __restrict__ moP) {
  const int i = blockIdx.x * 256 + threadIdx.x;
  if (i < Cout * WROW) {
    const int o = i / WROW, k = i - o * WROW;
    wP[(k >> 1) * (2 * Cout) + o * 2 + (k & 1)] = conv_w[i];
  } else {
    const int j = i - Cout * WROW;           // < 64*64
    const int o = j >> 6, c = j & 63;
    moP[(c >> 1) * (2 * Cout) + o * 2 + (c & 1)] = mlp_out_w[j];
  }
}

// ---------------- Kernel 1: brute-force KNN, top-20 ascending d2 ----------------
__global__ void __launch_bounds__(256) knn_kernel(const float* __restrict__ x,
                                                  int* __restrict__ idx_out) {
  __shared__ float spts[Nn * 3];
  const int b = blockIdx.x >> 3;            // 8 blocks per batch
  const int qbase = (blockIdx.x & 7) << 8;
  for (int i = threadIdx.x; i < Nn * 3; i += 256) {
    int n = i / 3, d = i - 3 * n;
    spts[i] = x[(b * 3 + d) * Nn + n];
  }
  __syncthreads();
  const int q = qbase + threadIdx.x;
  const float qx = spts[q*3+0], qy = spts[q*3+1], qz = spts[q*3+2];
  const float qsq = qx*qx + qy*qy + qz*qz;
  float bd[Kk]; int bi[Kk];
#pragma unroll
  for (int k = 0; k < Kk; ++k) { bd[k] = 3.4e38f; bi[k] = 0; }
  for (int m = 0; m < Nn; ++m) {
    const float mx = spts[m*3+0], my = spts[m*3+1], mz = spts[m*3+2];
    const float msq = mx*mx + my*my + mz*mz;
    float dd = qsq + msq - 2.0f*(qx*mx + qy*my + qz*mz);
    int ii = m;
#pragma unroll
    for (int k = 0; k < Kk; ++k) {          // stable bubble-insert (ties -> lower idx)
      if (dd < bd[k]) { float td=bd[k]; int ti=bi[k]; bd[k]=dd; bi[k]=ii; dd=td; ii=ti; }
    }
  }
  const long base = (long)(b * Nn + q) * Kk;
#pragma unroll
  for (int k = 0; k < Kk; ++k) idx_out[base + k] = bi[k];
}

// ---------------- Kernel 2: fused feats/perm/agg + f32 WMMA GEMMs ----------------
__global__ void __launch_bounds__(128)
paiconv_main(const float* __restrict__ x, const float* __restrict__ feature,
             const float* __restrict__ kernals, const float* __restrict__ one_padding,
             const float* __restrict__ mlp_w, const float* __restrict__ mlp_b,
             const float* __restrict__ wP, const float* __restrict__ conv_b,
             const float* __restrict__ moP, const float* __restrict__ mlp_out_b,
             const int* __restrict__ knn_idx, float* __restrict__ preout) {
  __shared__ float sPerm[MT*Kk*Ks];   // 6400 f  [p][k][m]
  __shared__ float sF7 [MT*Kk*8];     // 2560 f  [p][k][{rep3,rel3,dis,pad}]
  __shared__ float sF  [MT*CH*Kk];    // 2560 f  feats chunk [p][cl][k] / scratch / featT
  __shared__ float sAgg[MT*KCH];      // 2560 f  A operand [p][K]
  __shared__ int   sIdx[MT*Kk];       // 320 i
  const int tid = threadIdx.x;
  const int p0  = blockIdx.x * MT;
  const int b0  = p0 >> 11;           // whole block is one batch (MT | Nn)
  const int n0  = p0 & (Nn - 1);

  // phase 0a: gather neighbor coords into sF scratch ([pair][3])
  for (int t = tid; t < MT*Kk; t += 128) {
    const int j = knn_idx[(long)(p0 + t / Kk) * Kk + (t % Kk)];
    sIdx[t] = j;
    sF[t*3+0] = x[(b0*3+0)*Nn + j];
    sF[t*3+1] = x[(b0*3+1)*Nn + j];
    sF[t*3+2] = x[(b0*3+2)*Nn + j];
  }
  __syncthreads();
  // phase 0b: feats7 = [repeat(3), rel(3), dis]
  for (int t = tid; t < MT*Kk; t += 128) {
    const int p = t / Kk;
    const float r0x = sF[(p*Kk)*3+0], r0y = sF[(p*Kk)*3+1], r0z = sF[(p*Kk)*3+2];
    const float rx = sF[t*3+0]-r0x, ry = sF[t*3+1]-r0y, rz = sF[t*3+2]-r0z;
    float* f7 = &sF7[t*8];
    f7[0]=r0x; f7[1]=r0y; f7[2]=r0z; f7[3]=rx; f7[4]=ry; f7[5]=rz;
    f7[6]=sqrtf(rx*rx+ry*ry+rz*rz+1e-12f); f7[7]=0.f;
  }
  __syncthreads();
  // phase 1: perm (relu -> colnorm -> square -> colnorm -> threshold), per (p,m) column
  for (int t = tid; t < MT*Ks; t += 128) {
    const int p = t / Ks, m = t - p*Ks;
    float s1 = 0.f;
#pragma unroll
    for (int k = 0; k < Kk; ++k) {
      const float* f7 = &sF7[(p*Kk+k)*8];
      float v = f7[3]*kernals[0*Ks+m] + f7[4]*kernals[1*Ks+m] + f7[5]*kernals[2*Ks+m]
              + one_padding[k*Ks+m];
      v = v > 0.f ? v : 0.f;
      sPerm[(p*Kk+k)*Ks+m] = v; s1 += v;
    }
    const float i1 = 1.f/(s1+1e-6f);
    float s2 = 0.f;
#pragma unroll
    for (int k = 0; k < Kk; ++k) {
      float v = sPerm[(p*Kk+k)*Ks+m]*i1; v = v*v;
      sPerm[(p*Kk+k)*Ks+m] = v; s2 += v;
    }
    const float i2 = 1.f/(s2+1e-6f);
#pragma unroll
    for (int k = 0; k < Kk; ++k) {
      const float v = sPerm[(p*Kk+k)*Ks+m]*i2;
      sPerm[(p*Kk+k)*Ks+m] = (v > 0.1f) ? v : 0.f;
    }
  }
  __syncthreads();

  const int lane  = tid & 31;
  const int wave  = tid >> 5;
  const int o0    = wave * 16;                 // each wave owns one 16-wide N tile
  const int nlane = lane & 15;                 // A row M / B column N
  const int khalf = (lane >> 4) << 1;          // 0 or 2 (K sub-pair per half-wave)
  const float* wlane  = wP  + (o0 + nlane) * 2;   // packed: contiguous b64 per lane
  const float* molane = moP + (o0 + nlane) * 2;
  v8f acc = {0.f,0.f,0.f,0.f,0.f,0.f,0.f,0.f};

  for (int cc = 0; cc < NCHUNK; ++cc) {
    // feats chunk: rows 0..63 gathered neighbor feats, rows 64..127 the 7->64 MLP
    for (int t = tid; t < MT*CH*Kk; t += 128) {
      const int p = t / (CH*Kk), rem = t - p*(CH*Kk);
      const int cl = rem / Kk, k = rem - cl*Kk;
      float val;
      if (cc < 8) {
        const int c = cc*CH + cl;
        val = feature[((b0*Cin + c)*Nn) + sIdx[p*Kk+k]];
      } else {
        const int c2 = (cc-8)*CH + cl;
        const float* f7 = &sF7[(p*Kk+k)*8];
        const float* w  = &mlp_w[c2*7];
        val = mlp_b[c2] + w[0]*f7[0]+w[1]*f7[1]+w[2]*f7[2]
                        + w[3]*f7[3]+w[4]*f7[4]+w[5]*f7[5]+w[6]*f7[6];
      }
      sF[(p*CH+cl)*Kk + k] = val;
    }
    __syncthreads();
    // agg chunk = feats @ perm  -> A operand [p][cl*20+m]
    for (int t = tid; t < MT*CH*Ks; t += 128) {
      const int p = t / (CH*Ks), rem = t - p*(CH*Ks);
      const int cl = rem / Ks, m = rem - cl*Ks;
      float a = 0.f;
#pragma unroll
      for (int k = 0; k < Kk; ++k)
        a += sF[(p*CH+cl)*Kk+k] * sPerm[(p*Kk+k)*Ks+m];
      sAgg[p*KCH + cl*Ks + m] = a;
    }
    __syncthreads();
    // prefetch next chunk's packed weights into cache (global_prefetch_b8)
    if (cc + 1 < NCHUNK)
      __builtin_prefetch(wlane + (((cc + 1) * KCH) >> 1) * (2*Cout), 0, 3);
    // f32 WMMA accumulate over this chunk's 160 K values (40 x K=4)
    const int kbase = cc * KCH;
    const float* arow = &sAgg[nlane * KCH];
#pragma unroll 4
    for (int kk = 0; kk < KCH; kk += 4) {
      const v2f a  = *(const v2f*)&arow[kk + khalf];
      const v2f bf = *(const v2f*)(wlane + (((kbase + kk + khalf) >> 1) * (2*Cout)));
      acc = __builtin_amdgcn_wmma_f32_16x16x4_f32(false, a, false, bf,
                                                  (short)0, acc, false, false);
    }
    __syncthreads();
  }

  // ---- mlp_out residual as a 16x64x64 WMMA GEMM (featT tile in LDS) ----
  {
    const int p = tid & 15;
    for (int c = tid >> 4; c < Cin; c += 8)          // coalesced: consecutive p -> consecutive n
      sF[p*Cin + c] = feature[((b0*Cin + c)*Nn) + n0 + p];
  }
  __syncthreads();
#pragma unroll 4
  for (int kk = 0; kk < Cin; kk += 4) {
    const v2f a  = *(const v2f*)&sF[nlane*Cin + kk + khalf];
    const v2f bf = *(const v2f*)(molane + (((kk + khalf) >> 1) * (2*Cout)));
    acc = __builtin_amdgcn_wmma_f32_16x16x4_f32(false, a, false, bf,
                                                (short)0, acc, false, false);
  }

  // epilogue: biases, store pre-BN result
#pragma unroll
  for (int r = 0; r < 8; ++r) {
    const int row = r + ((lane >> 4) << 3);    // C/D layout: M = r (+8 for upper half)
    const int n = n0 + row;
    const int o = o0 + nlane;
    const float v = acc[r] + conv_b[o] + mlp_out_b[o];
    preout[((b0*Cout + o)*Nn) + n] = v;
  }
}

// ---------------- Kernel 3: per-channel batch stats (biased var) ----------------
__global__ void __launch_bounds__(256) bn_stats_kernel(const float* __restrict__ pre,
                                                       float* __restrict__ stats) {
  const int o = blockIdx.x;
  __shared__ float ss[256], sq[256];
  float s = 0.f, s2 = 0.f;
  for (int i = threadIdx.x; i < Bb*Nn; i += 256) {
    const int b = i >> 11, n = i & (Nn-1);
    const float v = pre[((b*Cout + o)*Nn) + n];
    s += v; s2 += v*v;
  }
  ss[threadIdx.x] = s; sq[threadIdx.x] = s2;
  __syncthreads();
  for (int off = 128; off > 0; off >>= 1) {
    if (threadIdx.x < off) { ss[threadIdx.x] += ss[threadIdx.x+off];
                             sq[threadIdx.x] += sq[threadIdx.x+off]; }
    __syncthreads();
  }
  if (threadIdx.x == 0) {
    const float inv = 1.0f / (float)(Bb*Nn);
    const float mean = ss[0]*inv;
    stats[o]        = mean;
    stats[Cout + o] = sq[0]*inv - mean*mean;
  }
}

// ---------------- Kernel 4: apply BN in place ----------------
__global__ void __launch_bounds__(256) bn_apply_kernel(float* __restrict__ out,
                                                       const float* __restrict__ stats,
                                                       const float* __restrict__ gamma,
                                                       const float* __restrict__ beta) {
  const int i = blockIdx.x*256 + threadIdx.x;
  const int o = (i >> 11) & (Cout-1);
  const float mean = stats[o], var = stats[Cout+o];
  out[i] = (out[i]-mean)*rsqrtf(var+1e-5f)*gamma[o]+beta[o];
}

extern "C" void kernel_launch(void* const* d_in, const int* in_sizes, int n_in,
                              void* d_out, int out_size, void* d_ws, size_t ws_size,
                              hipStream_t stream) {
  const float* x       = (const float*)d_in[0];
  const float* feature = (const float*)d_in[1];
  const float* kernals = (const float*)d_in[2];
  const float* one_pad = (const float*)d_in[3];
  const float* mlp_w   = (const float*)d_in[4];
  const float* mlp_b   = (const float*)d_in[5];
  const float* conv_w  = (const float*)d_in[6];
  const float* conv_b  = (const float*)d_in[7];
  const float* mo_w    = (const float*)d_in[8];
  const float* mo_b    = (const float*)d_in[9];
  const float* gamma   = (const float*)d_in[10];
  const float* beta    = (const float*)d_in[11];
  float* out = (float*)d_out;

  // ws layout: idx ints | packed conv_w | packed mlp_out_w | stats
  char* ws = (char*)d_ws;
  int*   idx  = (int*)ws;                                  ws += (size_t)Bb*Nn*Kk*sizeof(int);
  float* wPp  = (float*)ws;                                ws += (size_t)Cout*WROW*sizeof(float);
  float* moPp = (float*)ws;                                ws += (size_t)Cout*Cin*sizeof(float);
  float* stat = (float*)ws;

  pack_weights_kernel<<<(Cout*WROW + Cout*Cin)/256, 256, 0, stream>>>(conv_w, mo_w, wPp, moPp);
  knn_kernel<<<Bb*(Nn/256), 256, 0, stream>>>(x, idx);
  paiconv_main<<<(Bb*Nn)/MT, 128, 0, stream>>>(x, feature, kernals, one_pad,
                                               mlp_w, mlp_b, wPp, conv_b,
                                               moPp, mo_b, idx, out);
  bn_stats_kernel<<<Cout, 256, 0, stream>>>(out, stat);
  bn_apply_kernel<<<(Bb*Cout*Nn)/256, 256, 0, stream>>>(out, stat, gamma, beta);
}